// GraphAttentionLayer_11115375362362
// MI455X (gfx1250) — compile-verified
//
#include <hip/hip_runtime.h>

#define BB 8
#define NN 2048
#define FF 256
#define ALPHA 0.2f

typedef __attribute__((ext_vector_type(2))) float v2f;
typedef __attribute__((ext_vector_type(8))) float v8f;

// ---------------------------------------------------------------------------
// Kernel 1: Wh[b,n,o] = sum_f h[b,n,f] * W[f,o]   (f32 WMMA 16x16x4)
// One wave per 16x16 output tile; K swept 4 at a time.
// ---------------------------------------------------------------------------
__global__ void __launch_bounds__(256) gat_wh_gemm(const float* __restrict__ h,
                                                   const float* __restrict__ W,
                                                   float* __restrict__ Wh) {
  int wid  = blockIdx.x * 8 + (threadIdx.x >> 5);
  int lane = threadIdx.x & 31;
  int b    = wid >> 11;            // 128 row-tiles * 16 col-tiles = 2048 tiles/batch
  int rem  = wid & 2047;
  int row0 = (rem >> 4) << 4;
  int col0 = (rem & 15) << 4;

  int m16 = lane & 15;             // row (A) / col (B,C,D) within tile
  int kk0 = (lane >> 4) << 1;      // K sub-offset {0,2} per lane half

  const float* hrow = h + ((size_t)(b * NN + row0 + m16)) * FF;
  const float* wcol = W + (col0 + m16);

  v8f c = {};
  #pragma unroll 4
  for (int k = 0; k < FF; k += 4) {
    v2f a = *(const v2f*)(hrow + k + kk0);          // A: K=k+kk0, k+kk0+1 (contig)
    v2f bv;
    bv.x = wcol[(size_t)(k + kk0) * FF];            // B: rows of W, same K slots
    bv.y = wcol[(size_t)(k + kk0 + 1) * FF];
    c = __builtin_amdgcn_wmma_f32_16x16x4_f32(false, a, false, bv,
                                              (short)0, c, false, false);
  }

  // C/D layout: VGPR r -> row r (lanes 0-15) / row r+8 (lanes 16-31), col = lane&15
  int rbase = (lane >> 4) << 3;
  float* op = Wh + ((size_t)(b * NN + row0 + rbase)) * FF + col0 + m16;
  #pragma unroll
  for (int r = 0; r < 8; ++r)
    op[(size_t)r * FF] = c[r];
}

// ---------------------------------------------------------------------------
// Kernel 2: LDS-tiled transpose  WhT[b,c,n] = Wh[b,n,c]
// Makes the attention kernel's B-operand K-pairs contiguous (b64 loads).
// ---------------------------------------------------------------------------
__global__ void __launch_bounds__(256) gat_transpose(const float* __restrict__ Wh,
                                                     float* __restrict__ WhT) {
  __shared__ float tile[32][33];                   // +1 pad: no bank conflicts
  int blk = blockIdx.x;                            // B * (N/32) * (F/32) = 4096
  int b   = blk >> 9;                              // 512 tiles per batch
  int r   = blk & 511;
  int n0  = (r >> 3) << 5;                         // 64 n-tiles
  int c0  = (r & 7) << 5;                          // 8 c-tiles
  int tx  = threadIdx.x & 31;
  int ty  = threadIdx.x >> 5;                      // 0..7

  const float* src = Wh + ((size_t)(b * NN + n0 + ty)) * FF + c0 + tx;
  #pragma unroll
  for (int k = 0; k < 32; k += 8)
    tile[ty + k][tx] = src[(size_t)k * FF];
  __syncthreads();
  float* dst = WhT + ((size_t)(b * FF + c0 + ty)) * NN + n0 + tx;
  #pragma unroll
  for (int k = 0; k < 32; k += 8)
    dst[(size_t)k * NN] = tile[tx][ty + k];
}

// ---------------------------------------------------------------------------
// Kernel 3: s1[b,n] = Wh[b,n,:].a1 ; s2[b,n] = Wh[b,n,:].a2
// ---------------------------------------------------------------------------
__global__ void __launch_bounds__(256) gat_scores(const float* __restrict__ Wh,
                                                  const float* __restrict__ a,
                                                  float* __restrict__ s1,
                                                  float* __restrict__ s2) {
  int idx = blockIdx.x * 256 + threadIdx.x;        // 0 .. B*N-1
  const float* row = Wh + (size_t)idx * FF;
  float acc1 = 0.f, acc2 = 0.f;
  #pragma unroll 8
  for (int o = 0; o < FF; ++o) {
    float v = row[o];
    acc1 += v * a[o];
    acc2 += v * a[FF + o];
  }
  s1[idx] = acc1;
  s2[idx] = acc2;
}

// ---------------------------------------------------------------------------
// Kernel 4: fused masked leaky-relu logits -> online softmax -> P @ Wh -> ELU
// Workgroup = 256 thr = 8 waves, covers 64 rows. Wave w: row group w>>1
// (16 rows), column half w&1 (128 cols = 8 accumulator tiles). Streams j in
// chunks of 32 with flash-style online softmax; logit/exp work is computed
// exactly where the WMMA A-operand slots need it (row = lane&15, K-slots =
// (lane>=16)*2 + 4t), row reductions via one xor-16 shuffle.
// ---------------------------------------------------------------------------
__global__ void __launch_bounds__(256) gat_attn(const float* __restrict__ WhT,
                                                const int*   __restrict__ adj,
                                                const float* __restrict__ s1,
                                                const float* __restrict__ s2,
                                                float* __restrict__ out) {
  int blk  = blockIdx.x;           // B * (N/64) = 256
  int b    = blk >> 5;
  int i0   = (blk & 31) << 6;      // 64 rows per workgroup
  int w    = threadIdx.x >> 5;
  int lane = threadIdx.x & 31;
  int g    = w >> 1;               // row group 0..3
  int ch   = w & 1;                // column half 0..1
  int m16  = lane & 15;
  int kk0  = (lane >> 4) << 1;     // K sub-offset {0,2}
  int i    = i0 + (g << 4) + m16;  // global row handled by this lane (A side)
  int cb   = ch << 7;              // column base: 0 or 128

  const float* whtb   = WhT + (size_t)b * FF * NN;
  const float* s2b    = s2 + b * NN;
  const int*   adjrow = adj + (size_t)i * NN;
  float s1i = s1[b * NN + i];

  // one per-lane address; the 8 column streams are reached via immediate
  // offsets ct*16*NN*4 (<= 917KB, fits signed 24-bit IOFFSET)
  const float* pbase = whtb + (size_t)(cb + m16) * NN + kk0;

  float mrun = -INFINITY, lrun = 0.f;
  v8f acc[8];
  #pragma unroll
  for (int ct = 0; ct < 8; ++ct) acc[ct] = (v8f){};

  for (int j0 = 0; j0 < NN; j0 += 32) {
    if (j0 + 32 < NN) {                            // next-chunk prefetch
      __builtin_prefetch(adjrow + j0 + 32, 0, 1);  // -> global_prefetch_b8
      __builtin_prefetch(s2b + j0 + 32, 0, 1);
    }
    // --- logits for exactly the 16 (i,j) pairs this lane's A-slots need ---
    float2 pv[8];
    float tmax = -INFINITY;
    #pragma unroll
    for (int u = 0; u < 8; ++u) {                  // u = tau*4 + t
      int jc = j0 + ((u >> 2) << 4) + ((u & 3) << 2) + kk0;
      float2 sv = *(const float2*)(s2b + jc);
      int2   av = *(const int2*)(adjrow + jc);
      float e0 = s1i + sv.x; e0 = e0 > 0.f ? e0 : ALPHA * e0;
      float e1 = s1i + sv.y; e1 = e1 > 0.f ? e1 : ALPHA * e1;
      if (!((av.x > 0) | (i == jc)))     e0 = -INFINITY;
      if (!((av.y > 0) | (i == jc + 1))) e1 = -INFINITY;
      pv[u].x = e0; pv[u].y = e1;
      tmax = fmaxf(tmax, fmaxf(e0, e1));
    }
    // rows are split between lane and lane^16: one xor-shuffle closes the row
    tmax = fmaxf(tmax, __shfl_xor(tmax, 16, 32));
    float mnew  = fmaxf(mrun, tmax);
    float scale = (mnew > mrun) ? __expf(mrun - mnew) : 1.0f;  // -inf safe

    float psum = 0.f;
    #pragma unroll
    for (int u = 0; u < 8; ++u) {
      float p0 = (pv[u].x == -INFINITY) ? 0.f : __expf(pv[u].x - mnew);
      float p1 = (pv[u].y == -INFINITY) ? 0.f : __expf(pv[u].y - mnew);
      pv[u].x = p0; pv[u].y = p1;
      psum += p0 + p1;
    }
    psum += __shfl_xor(psum, 16, 32);
    lrun = lrun * scale + psum;
    mrun = mnew;

    #pragma unroll
    for (int ct = 0; ct < 8; ++ct)
      #pragma unroll
      for (int r = 0; r < 8; ++r) acc[ct][r] *= scale;

    // --- P(16x32-chunk) @ Wh(32x128) : 8 K-steps x 8 column tiles of WMMA ---
    #pragma unroll
    for (int u = 0; u < 8; ++u) {
      int kof = j0 + ((u >> 2) << 4) + ((u & 3) << 2);   // K window base
      v2f aop; aop.x = pv[u].x; aop.y = pv[u].y;
      #pragma unroll
      for (int ct = 0; ct < 8; ++ct) {
        // B pair (rows k,k+1 @ col) contiguous in WhT -> single b64 load
        v2f bop = *(const v2f*)(pbase + (size_t)(ct << 4) * NN + kof);
        acc[ct] = __builtin_amdgcn_wmma_f32_16x16x4_f32(false, aop, false, bop,
                                                        (short)0, acc[ct],
                                                        false, false);
      }
    }
  }

  // --- epilogue: divide by row sum (0 if fully masked, like nan_to_num), ELU ---
  int rbase = (lane >> 4) << 3;
  #pragma unroll
  for (int r = 0; r < 8; ++r) {
    int rowm  = r + rbase;                         // 0..15 within row group
    float lr  = __shfl(lrun, rowm, 32);            // lanes 0-15 hold the sums
    float inv = (lr > 0.f) ? 1.0f / lr : 0.f;
    float* op = out + ((size_t)(b * NN + i0 + (g << 4) + rowm)) * FF + cb + m16;
    #pragma unroll
    for (int ct = 0; ct < 8; ++ct) {
      float v = acc[ct][r] * inv;
      v = v > 0.f ? v : (__expf(v) - 1.0f);        // ELU
      op[ct << 4] = v;
    }
  }
}

// ---------------------------------------------------------------------------
extern "C" void kernel_launch(void* const* d_in, const int* in_sizes, int n_in,
                              void* d_out, int out_size, void* d_ws, size_t ws_size,
                              hipStream_t stream) {
  const float* h   = (const float*)d_in[0];   // (B, N, Fin) f32
  const int*   adj = (const int*)  d_in[1];   // (N, N) i32
  const float* W   = (const float*)d_in[2];   // (Fin, Fout) f32
  const float* a   = (const float*)d_in[3];   // (2*Fout, 1) f32
  float* out = (float*)d_out;                 // (B, N, Fout) f32

  // workspace: Wh (16 MB) + WhT (16 MB) + s1/s2 (64 KB each)
  float* Wh  = (float*)d_ws;
  float* WhT = Wh + (size_t)BB * NN * FF;
  float* s1  = WhT + (size_t)BB * NN * FF;
  float* s2  = s1 + (size_t)BB * NN;

  // 1) Wh = h @ W : 16384 tiles, 8 waves/block -> 2048 blocks
  gat_wh_gemm<<<(BB * (NN / 16) * (FF / 16)) / 8, 256, 0, stream>>>(h, W, Wh);
  // 2) transpose for contiguous B-operand access in the attention GEMM
  gat_transpose<<<BB * (NN / 32) * (FF / 32), 256, 0, stream>>>(Wh, WhT);
  // 3) attention source/target scores
  gat_scores<<<(BB * NN) / 256, 256, 0, stream>>>(Wh, a, s1, s2);
  // 4) fused masked softmax + P@Wh + ELU : one block per (b, 64-row tile)
  gat_attn<<<BB * (NN / 64), 256, 0, stream>>>(WhT, adj, s1, s2, out);
}